// backprojector_76484777607439
// MI455X (gfx1250) — compile-verified
//
#include <hip/hip_runtime.h>
#include <math.h>

// Problem constants (match reference)
#define VIEWS 512
#define DETS  512
#define IMG_H 256
#define IMG_W 256

typedef __attribute__((ext_vector_type(2))) float v2f;
typedef __attribute__((ext_vector_type(4))) float v4f;
typedef __attribute__((ext_vector_type(8))) float v8f;

#if defined(__HIP_DEVICE_COMPILE__) && __has_builtin(__builtin_amdgcn_wmma_f32_16x16x4_f32)
#define WMMA_ACC(acc, a, b)                                                   \
    acc = __builtin_amdgcn_wmma_f32_16x16x4_f32(false, (a), false, (b),       \
                                                (short)0, (acc), false, false)
#define RFL(x) __builtin_amdgcn_readfirstlane(x)
#else
#define WMMA_ACC(acc, a, b) do { (acc)[0] += (a).x * (b).x; } while (0) // host stub
#define RFL(x) (x)
#endif

__device__ __forceinline__ float hatw(float f) {
    return fmaxf(0.0f, 1.0f - fabsf(f));   // linear-interp hat function
}

// Backprojection as hat-function GEMM on the f32 WMMA pipe:
//   D(16 px x 16 cols, cols 0..3 = batches) +=
//       A(16 px x 4 dets of hat weights) x B(4 dets x 16 cols of proj)
// accumulated over each view's <=12-detector window and over all 512 views.
//
// Wave (32 lanes) owns one 16-pixel x-strip at fixed y, all 4 batches:
//  A 16x4 f32 : lane l -> M=l&15, K=(l<16?0:2)+{v0,v1}
//  B  4x16 f32: lane l -> N=l&15, K=(l<16?0:2)+{v0,v1}  (consecutive dets!)
//  D 16x16 f32: VGPR r -> M=r+(l<16?0:8), N=l&15 -> 8 consecutive x per lane
__global__ __launch_bounds__(128, 4)
void backproject_wmma(const float* __restrict__ proj,
                      const float* __restrict__ opt,
                      float* __restrict__ out)
{
    __shared__ float2 s_trig[VIEWS];   // {cos,sin} * dImg/dDet per view

    const float dImg = opt[0];
    const float dDet = opt[1];
    const float ang0 = opt[2];
    const float dAng = opt[3];
    const float kk   = dImg / dDet;

    for (int i = (int)threadIdx.x; i < VIEWS; i += 128) {
        float s, c;
        sincosf(fmaf((float)i, dAng, ang0), &s, &c);
        s_trig[i] = make_float2(c * kk, s * kk);
    }
    __syncthreads();

    const int lane = threadIdx.x & 31;
    const int wv   = threadIdx.x >> 5;
    const int m    = lane & 15;          // pixel row (A-M) == batch col (B-N)
    const int klo  = (lane >> 4) << 1;   // K pair base: 0 (low lanes) / 2 (high)
    const int y    = blockIdx.y * 4 + wv;
    const int x0   = blockIdx.x * 16;

    const float xm  = (float)(x0 + m) - 0.5f * (IMG_W - 1);
    const float x0c = (float)x0        - 0.5f * (IMG_W - 1);
    const float yc  = 0.5f * (IMG_H - 1) - (float)y;
    const float uc  = 0.5f * (DETS - 1);

    const bool     bcol = (m < 4);
    const unsigned bofs = (bcol ? (unsigned)m : 0u) * (unsigned)(VIEWS * DETS);

    v8f acc0 = {}, acc1 = {};            // two chains -> half the WMMA RAW depth

    // ---- stage view v: SGPR window math + (fast path) 6-load clause ----
    // dmin/dmax are wave-uniform: readfirstlane them into SGPRs so the
    // ok-test and tail bound become s_cmp/s_cbranch, not v_cmpx EXEC churn.
    auto stage = [&](int v, float& su, int& sdmin, int& sdmax, bool& sok,
                     v2f& sb0, v2f& sb1, v2f& sb2) {
        const float2 t  = s_trig[v];               // ds_load_b64, uniform
        const float sx = t.x, sy = t.y;
        const float bse = fmaf(sy, yc, uc);
        su = fmaf(sx, xm, bse);
        const float u0  = fmaf(sx, x0c, bse);
        const float spn = sx * 15.0f;
        sdmin = RFL((int)floorf(u0 + fminf(spn, 0.0f)));
        sdmax = RFL((int)floorf(u0 + fmaxf(spn, 0.0f)) + 1);
        sok   = (sdmin >= 0) & (sdmin + 11 < DETS);   // scalar branch condition
        sb0 = v2f{}; sb1 = v2f{}; sb2 = v2f{};
        if (sok) {                                    // uniform (SGPR) branch
            if (bcol) {                               // one EXEC region per view
                const unsigned o = bofs + (unsigned)(v * DETS)
                                        + (unsigned)(sdmin + klo);
                sb0.x = proj[o];     sb0.y = proj[o + 1];   // imm offsets ->
                sb1.x = proj[o + 4]; sb1.y = proj[o + 5];   //  one s_clause
                sb2.x = proj[o + 8]; sb2.y = proj[o + 9];
                const int vp = min(v + 8, VIEWS - 1);       // L1 lookahead
                __builtin_prefetch(proj + bofs + (unsigned)(vp * DETS)
                                        + (unsigned)(sdmin + klo), 0, 1);
            }
        }
    };

    // ---- consume view v: weights + 3 WMMAs (fast) or masked tail (cold) ----
    auto consume = [&](int v, float cu, int cdmin, int cdmax, bool cok,
                       v2f cb0, v2f cb1, v2f cb2) {
        int d0t;
        if (cok) {                                    // uniform; EXEC stays full
            const float f0 = cu - (float)(cdmin + klo);
            v2f a0; a0.x = hatw(f0);          a0.y = hatw(f0 - 1.0f);
            v2f a1; a1.x = hatw(f0 - 4.0f);   a1.y = hatw(f0 - 5.0f);
            v2f a2; a2.x = hatw(f0 - 8.0f);   a2.y = hatw(f0 - 9.0f);
            WMMA_ACC(acc0, a0, cb0);
            WMMA_ACC(acc1, a1, cb1);
            WMMA_ACC(acc0, a2, cb2);
            d0t = cdmin + 12;
        } else {
            d0t   = max(cdmin, -1);
            cdmax = min(cdmax, DETS);
        }
        // General clamped/masked path: never taken for the reference options.
        // Bounds are SGPRs -> scalar loop control.
        for (int d0 = d0t; d0 <= cdmax; d0 += 4) {
            const int dk0 = d0 + klo, dk1 = dk0 + 1;
            const float f = cu - (float)dk0;
            float w0 = hatw(f), w1 = hatw(f - 1.0f);
            if ((unsigned)dk0 >= DETS) w0 = 0.0f;
            if ((unsigned)dk1 >= DETS) w1 = 0.0f;
            v2f a; a.x = w0; a.y = w1;
            v2f b = v2f{};
            if (bcol) {
                const unsigned rb = bofs + (unsigned)(v * DETS);
                b.x = proj[rb + (unsigned)min(max(dk0, 0), DETS - 1)];
                b.y = proj[rb + (unsigned)min(max(dk1, 0), DETS - 1)];
            }
            WMMA_ACC(acc1, a, b);
        }
    };

    // ---- two-view-deep software pipeline: even if the scheduler sinks one
    //      stage's loads below the WMMAs, a full stage stays in flight ----
    float u_c;  int dmin_c,  dmax_c;  bool ok_c;  v2f b0_c,  b1_c,  b2_c;
    float u_n1; int dmin_n1, dmax_n1; bool ok_n1; v2f b0_n1, b1_n1, b2_n1;
    stage(0, u_c,  dmin_c,  dmax_c,  ok_c,  b0_c,  b1_c,  b2_c);
    stage(1, u_n1, dmin_n1, dmax_n1, ok_n1, b0_n1, b1_n1, b2_n1);
    for (int v = 0; v < VIEWS - 2; ++v) {
        float u_n2; int dmin_n2, dmax_n2; bool ok_n2; v2f b0_n2, b1_n2, b2_n2;
        stage(v + 2, u_n2, dmin_n2, dmax_n2, ok_n2, b0_n2, b1_n2, b2_n2);
        consume(v, u_c, dmin_c, dmax_c, ok_c, b0_c, b1_c, b2_c);
        u_c  = u_n1;  dmin_c  = dmin_n1;  dmax_c  = dmax_n1;  ok_c  = ok_n1;
        b0_c = b0_n1; b1_c = b1_n1; b2_c = b2_n1;
        u_n1 = u_n2;  dmin_n1 = dmin_n2;  dmax_n1 = dmax_n2;  ok_n1 = ok_n2;
        b0_n1 = b0_n2; b1_n1 = b1_n2; b2_n1 = b2_n2;
    }
    consume(VIEWS - 2, u_c,  dmin_c,  dmax_c,  ok_c,  b0_c,  b1_c,  b2_c);
    consume(VIEWS - 1, u_n1, dmin_n1, dmax_n1, ok_n1, b0_n1, b1_n1, b2_n1);

    // ---- D -> img: lanes N=m<4 store 8 consecutive x (two b128 stores) ----
    if (bcol) {
        const v8f acc = acc0 + acc1;
        const int mh  = (lane >> 4) * 8;
        float* o = out + (size_t)m * (IMG_H * IMG_W) + (size_t)y * IMG_W + x0 + mh;
        v4f lo = { acc[0] * dAng, acc[1] * dAng, acc[2] * dAng, acc[3] * dAng };
        v4f hi = { acc[4] * dAng, acc[5] * dAng, acc[6] * dAng, acc[7] * dAng };
        *reinterpret_cast<v4f*>(o)     = lo;   // 16B aligned: x0%16==0, mh%8==0
        *reinterpret_cast<v4f*>(o + 4) = hi;
    }
}

extern "C" void kernel_launch(void* const* d_in, const int* in_sizes, int n_in,
                              void* d_out, int out_size, void* d_ws, size_t ws_size,
                              hipStream_t stream)
{
    const float* proj = (const float*)d_in[0];   // (4,1,512,512) f32
    const float* opt  = (const float*)d_in[1];   // [dImg, dDet, ang0, dAng]
    float*       out  = (float*)d_out;           // (4,1,256,256) f32

    dim3 grid(IMG_W / 16, IMG_H / 4);
    dim3 block(128);                             // 4 wave32 waves / block
    hipLaunchKernelGGL(backproject_wmma, grid, block, 0, stream, proj, opt, out);
}